// GCN_19937238188789
// MI455X (gfx1250) — compile-verified
//
#include <hip/hip_runtime.h>

// GCN 2-layer: (X@W1+b1) -> SpMM -> ReLU -> (@W2+b2) -> SpMM
// fp32 WMMA (V_WMMA_F32_16X16X4_F32) for the dense GEMMs,
// wave-per-edge gather + hw f32 atomics for the COO SpMM.

typedef __attribute__((ext_vector_type(2))) float v2f;
typedef __attribute__((ext_vector_type(8))) float v8f;

#define GCN_HIDDEN 128
#define GCN_KCHUNK 64

// ---------------------------------------------------------------------------
// One-shot repack of W [K][128] (row-major) into fragment-major order so that
// a B-operand fragment is one contiguous 8-byte ds_load_b64 per lane:
//   Wsw[ (ks*8 + t)*64 + lane*2 + j ] = W[ ks*4 + ((lane>>4)<<1) + j ][ t*16 + (lane&15) ]
// (ks = k/4 group, t = 16-col tile, lane = wave lane, j = element 0/1)
// Matches the CDNA5 WMMA 4x16 B layout: VGPR0 = rows K0/K2 per half-wave,
// VGPR1 = rows K1/K3.
// ---------------------------------------------------------------------------
__global__ __launch_bounds__(256)
void gcn_wswizzle_kernel(const float* __restrict__ W, float* __restrict__ Wsw,
                         int n_elems)
{
    int idx = blockIdx.x * 256 + threadIdx.x;
    if (idx >= n_elems) return;
    int grp  = idx >> 6;          // ks*8 + t
    int rem  = idx & 63;          // lane*2 + j
    int lane = rem >> 1;
    int j    = rem & 1;
    int ks   = grp >> 3;
    int t    = grp & 7;
    int kofs = (lane >> 4) << 1;
    int ln   = lane & 15;
    int krow = ks * 4 + kofs + j;
    Wsw[idx] = W[krow * GCN_HIDDEN + t * 16 + ln];
}

// ---------------------------------------------------------------------------
// GEMM + bias (+ optional fused ReLU on the A operand):  H = act(X) @ Wsw + b
// Block = 256 threads = 8 waves. Block tile: 256 rows x 128 cols.
// Wave w owns TWO 16-row strips (rows w*16.. and w*16+128..), all 128 cols:
// the same 8 B fragments feed 16 WMMAs, halving LDS bytes per FLOP (the
// binding resource for K=4 fp32 WMMA) vs. one strip per wave.
// Swizzled W staged through LDS in K-chunks of 64 (32 KB, contiguous copy).
// NTA: non-temporal A loads (layer 1 streams X once; keep L2 for H).
// ---------------------------------------------------------------------------
template <int K, bool RELU, bool NTA>
__global__ __launch_bounds__(256)
void gcn_gemm_bias_kernel(const float* __restrict__ X,
                          const float* __restrict__ Wsw,
                          const float* __restrict__ bias,
                          float* __restrict__ H,
                          int n_rows)
{
    __shared__ float Wlds[GCN_KCHUNK * GCN_HIDDEN]; // 32 KB, fragment-major

    const int tid  = threadIdx.x;
    const int wave = tid >> 5;
    const int lane = tid & 31;
    const int ln   = lane & 15;          // col index within 16-wide tile
    const int kofs = (lane >> 4) << 1;   // 0 for lanes 0-15, 2 for lanes 16-31

    const int m0 = blockIdx.x * 256 + wave * 16;   // strip 0
    const int m1 = m0 + 128;                       // strip 1

    // Clamp A rows for the ragged last block (stores are guarded; duplicate
    // loads are harmless and keep EXEC all-ones for WMMA).
    int r0 = m0 + ln; if (r0 > n_rows - 1) r0 = n_rows - 1; if (r0 < 0) r0 = 0;
    int r1 = m1 + ln; if (r1 > n_rows - 1) r1 = n_rows - 1; if (r1 < 0) r1 = 0;
    const float* Xrow0 = X + (size_t)r0 * K + kofs;
    const float* Xrow1 = X + (size_t)r1 * K + kofs;
    const float* Bf    = Wlds + lane * 2;   // per-lane fragment base

    // Accumulators seeded with bias (bias depends only on N -> broadcast).
    v8f acc0[8], acc1[8];
#pragma unroll
    for (int t = 0; t < 8; ++t) {
        float bv = bias[t * 16 + ln];
        v8f a0 = {bv, bv, bv, bv, bv, bv, bv, bv};
        acc0[t] = a0;
        acc1[t] = a0;
    }

    for (int k0 = 0; k0 < K; k0 += GCN_KCHUNK) {
        __syncthreads();
        // Cooperative copy of one contiguous 32 KB chunk of Wsw into LDS.
        const float4* Wg = (const float4*)(Wsw + (size_t)k0 * GCN_HIDDEN);
        float4* Wl = (float4*)Wlds;
#pragma unroll
        for (int i = 0; i < (GCN_KCHUNK * GCN_HIDDEN / 4) / 256; ++i)
            Wl[tid + i * 256] = Wg[tid + i * 256];
        __syncthreads();

#pragma unroll
        for (int kk = 0; kk < GCN_KCHUNK; kk += 4) {
            // A fragments: {X[row][k+kofs], X[row][k+kofs+1]} for both strips.
            v2f a0, a1;
            if (NTA) {
                a0 = __builtin_nontemporal_load((const v2f*)(Xrow0 + k0 + kk));
                a1 = __builtin_nontemporal_load((const v2f*)(Xrow1 + k0 + kk));
            } else {
                a0 = *(const v2f*)(Xrow0 + k0 + kk);
                a1 = *(const v2f*)(Xrow1 + k0 + kk);
            }
            if (RELU) {
                a0.x = fmaxf(a0.x, 0.0f); a0.y = fmaxf(a0.y, 0.0f);
                a1.x = fmaxf(a1.x, 0.0f); a1.y = fmaxf(a1.y, 0.0f);
            }

            // 8 B fragments (conflict-free ds_load_b64 each, pair-mergeable
            // into ds_load_2addr_b64) feed 16 WMMAs.
            v2f bf[8];
#pragma unroll
            for (int t = 0; t < 8; ++t)
                bf[t] = *(const v2f*)(Bf + (((kk >> 2) * 8) + t) * 64);

#pragma unroll
            for (int t = 0; t < 8; ++t) {
                // Adjacent WMMAs share the B operand (operand-cache friendly).
                acc0[t] = __builtin_amdgcn_wmma_f32_16x16x4_f32(
                    false, a0, false, bf[t], (short)0, acc0[t], false, false);
                acc1[t] = __builtin_amdgcn_wmma_f32_16x16x4_f32(
                    false, a1, false, bf[t], (short)0, acc1[t], false, false);
            }
        }
    }

    // Store D: VGPR r -> row m + r (+8 for upper half-wave), col t*16 + ln.
#pragma unroll
    for (int t = 0; t < 8; ++t) {
#pragma unroll
        for (int r = 0; r < 8; ++r) {
            int mrow = r + ((lane < 16) ? 0 : 8);
            int ma = m0 + mrow;
            int mb = m1 + mrow;
            if (ma < n_rows)
                H[(size_t)ma * GCN_HIDDEN + t * 16 + ln] = acc0[t][r];
            if (mb < n_rows)
                H[(size_t)mb * GCN_HIDDEN + t * 16 + ln] = acc1[t][r];
        }
    }
}

// ---------------------------------------------------------------------------
// COO SpMM: out[row[e]] += vals[e] * H[col[e]]   (one wave per edge, d=128)
// 512 B coalesced row gather per wave; 4 hardware f32 atomics per lane.
// H (51.2 MB) is L2-resident on MI455X (192 MB L2) -> L2-bandwidth bound.
// Edge arrays are streamed once -> non-temporal to preserve L2 for H.
// ---------------------------------------------------------------------------
__global__ __launch_bounds__(256)
void gcn_spmm_kernel(const float* __restrict__ H,
                     const float* __restrict__ vals,
                     const int* __restrict__ row,
                     const int* __restrict__ col,
                     float* __restrict__ out,
                     int n_edges)
{
    const int e    = (int)((blockIdx.x * (unsigned)blockDim.x + threadIdx.x) >> 5);
    const int lane = threadIdx.x & 31;
    if (e >= n_edges) return;

    const int   r = __builtin_nontemporal_load(row + e);
    const int   c = __builtin_nontemporal_load(col + e);
    const float v = __builtin_nontemporal_load(vals + e);

    float4 h = ((const float4*)(H + (size_t)c * GCN_HIDDEN))[lane];
    float* dst = out + (size_t)r * GCN_HIDDEN + lane * 4;

    unsafeAtomicAdd(dst + 0, v * h.x);
    unsafeAtomicAdd(dst + 1, v * h.y);
    unsafeAtomicAdd(dst + 2, v * h.z);
    unsafeAtomicAdd(dst + 3, v * h.w);
}

// ---------------------------------------------------------------------------
extern "C" void kernel_launch(void* const* d_in, const int* in_sizes, int n_in,
                              void* d_out, int out_size, void* d_ws, size_t ws_size,
                              hipStream_t stream)
{
    const float* X    = (const float*)d_in[0];
    const float* W1   = (const float*)d_in[1];
    const float* b1   = (const float*)d_in[2];
    const float* W2   = (const float*)d_in[3];
    const float* b2   = (const float*)d_in[4];
    const float* vals = (const float*)d_in[5];
    const int*   row  = (const int*)d_in[6];
    const int*   col  = (const int*)d_in[7];
    float*       out  = (float*)d_out;

    const int n = in_sizes[0] / 256;  // N_NODES (IN_SIZE = 256)
    const int e = in_sizes[5];        // N_EDGES

    // Scratch layout: H ping-pong + swizzled weights.
    float* bufA = (float*)d_ws;                        // H0, then H2
    float* bufB = bufA + (size_t)n * GCN_HIDDEN;       // H1
    float* W1s  = bufB + (size_t)n * GCN_HIDDEN;       // 128 KB
    float* W2s  = W1s + 256 * GCN_HIDDEN;              // 64 KB

    const int gblocks = (n + 255) / 256;
    const int eblocks = (e + 7) / 8;   // 8 edge-waves per 256-thread block

    // Repack weights into WMMA B-fragment order (tiny, runs once per call).
    gcn_wswizzle_kernel<<<(256 * GCN_HIDDEN + 255) / 256, 256, 0, stream>>>(
        W1, W1s, 256 * GCN_HIDDEN);
    gcn_wswizzle_kernel<<<(128 * GCN_HIDDEN + 255) / 256, 256, 0, stream>>>(
        W2, W2s, 128 * GCN_HIDDEN);

    // Layer 1: H0 = X @ W1 + b1   (X streamed non-temporally)
    gcn_gemm_bias_kernel<256, false, true><<<gblocks, 256, 0, stream>>>(
        X, W1s, b1, bufA, n);

    // H1 = SpMM(A, H0)   (zero accumulator first; capture-safe memset)
    hipMemsetAsync(bufB, 0, (size_t)n * GCN_HIDDEN * sizeof(float), stream);
    gcn_spmm_kernel<<<eblocks, 256, 0, stream>>>(bufA, vals, row, col, bufB, e);

    // Layer 2: H2 = relu(H1) @ W2 + b2   (ReLU fused; H1 from L2, keep RT)
    gcn_gemm_bias_kernel<128, true, false><<<gblocks, 256, 0, stream>>>(
        bufB, W2s, b2, bufA, n);

    // out = SpMM(A, H2)
    hipMemsetAsync(out, 0, (size_t)n * GCN_HIDDEN * sizeof(float), stream);
    gcn_spmm_kernel<<<eblocks, 256, 0, stream>>>(bufA, vals, row, col, out, e);
}